// TokenChoiceTopKRouter_26113401160073
// MI455X (gfx1250) — compile-verified
//
#include <hip/hip_runtime.h>
#include <hip/hip_bf16.h>
#include <math.h>

#define TOKENS 16384
#define DIM    2048
#define NEXP   64
#define TOPK   8
#define CHUNK_TOKENS 128
#define NCHUNKS (TOKENS / CHUNK_TOKENS)      /* 128 */
#define NSEL (TOKENS * TOPK)                 /* 131072 */
#define NTILES (TOKENS / 16)                 /* 1024 */

typedef float v2f __attribute__((ext_vector_type(2)));
typedef float v8f __attribute__((ext_vector_type(8)));

// ---------------------------------------------------------------------------
// Kernel 1: partial logits via V_WMMA_F32_16X16X4_F32 with deterministic
// K-split. Wave gw -> tile = gw / nsplit, k-slice s = gw % nsplit.
// Each wave: 16 tokens x 64 experts (4 n-tile accumulators) over K/nsplit.
// A fragment (16x4 f32, 2 VGPRs): lanes 0-15 hold row M=lane, K={k,k+1};
// lanes 16-31 hold K={k+2,k+3} -> contiguous float2 load of x.
// B fragment (4x16 f32, 2 VGPRs): B[k][n] = W[n][k]; same half-lane K-pair
// selection -> contiguous float2 load of W's row.
// Partials written raw (no bias/sigmoid); reduced in the top-k kernel.
// ---------------------------------------------------------------------------
__global__ __launch_bounds__(256) void router_gemm_partial_kernel(
    const float* __restrict__ x, const float* __restrict__ W,
    float* __restrict__ partials, int nsplit)
{
    const int lane = threadIdx.x & 31;
    const int wave = threadIdx.x >> 5;
    const int half = lane >> 4;      // 0: K pair {0,1}, 1: K pair {2,3}
    const int lm   = lane & 15;

    const int gw   = blockIdx.x * 8 + wave;
    const int tile = gw / nsplit;              // 0..1023
    const int ks   = gw - tile * nsplit;       // 0..nsplit-1
    const int kn   = DIM / nsplit;
    const int k0   = ks * kn;
    const int row0 = tile * 16;

    const float* xrow = x + (size_t)(row0 + lm) * DIM + 2 * half + k0;
    const float* w0   = W + (size_t)( 0 + lm) * DIM + 2 * half + k0;
    const float* w1   = W + (size_t)(16 + lm) * DIM + 2 * half + k0;
    const float* w2   = W + (size_t)(32 + lm) * DIM + 2 * half + k0;
    const float* w3   = W + (size_t)(48 + lm) * DIM + 2 * half + k0;

    v8f c0 = {}, c1 = {}, c2 = {}, c3 = {};

    for (int k = 0; k < kn; k += 4) {
        v2f a  = *(const v2f*)(xrow + k);
        v2f b0 = *(const v2f*)(w0 + k);
        v2f b1 = *(const v2f*)(w1 + k);
        v2f b2 = *(const v2f*)(w2 + k);
        v2f b3 = *(const v2f*)(w3 + k);
        c0 = __builtin_amdgcn_wmma_f32_16x16x4_f32(false, a, false, b0, (short)0, c0, false, false);
        c1 = __builtin_amdgcn_wmma_f32_16x16x4_f32(false, a, false, b1, (short)0, c1, false, false);
        c2 = __builtin_amdgcn_wmma_f32_16x16x4_f32(false, a, false, b2, (short)0, c2, false, false);
        c3 = __builtin_amdgcn_wmma_f32_16x16x4_f32(false, a, false, b3, (short)0, c3, false, false);
    }

    // C/D layout: lanes 0-15 -> M = vgpr, lanes 16-31 -> M = vgpr + 8; N = lm.
    float* pbase = partials + (size_t)ks * TOKENS * NEXP;
    #pragma unroll
    for (int i = 0; i < 8; ++i) {
        const int t = row0 + i + 8 * half;
        float* prow = pbase + (size_t)t * NEXP;
        prow[ 0 + lm] = c0[i];
        prow[16 + lm] = c1[i];
        prow[32 + lm] = c2[i];
        prow[48 + lm] = c3[i];
    }
}

// ---------------------------------------------------------------------------
// Kernel 2: fixed-order partial reduction + bias + sigmoid, then per-token
// stable top-8 + per-chunk expert histogram. One thread per token; 128
// tokens per block (= one counting-sort chunk). Strict '>' scan over
// ascending expert ids reproduces the stable tie-break of
// argsort(-scores, stable): lowest expert index wins ties.
// ---------------------------------------------------------------------------
__global__ __launch_bounds__(128) void router_topk_kernel(
    const float* __restrict__ partials, const float* __restrict__ bias,
    int* __restrict__ sel, float* __restrict__ selsc,
    int* __restrict__ chunkCounts, int nsplit)
{
    __shared__ float s[CHUNK_TOKENS * 65];   // pad 64->65: conflict-free
    __shared__ int   hist[NEXP];

    const int tid = threadIdx.x;
    if (tid < NEXP) hist[tid] = 0;

    const size_t gbase = (size_t)blockIdx.x * CHUNK_TOKENS * NEXP;
    for (int idx = tid; idx < CHUNK_TOKENS * NEXP; idx += 128) {
        const int t = idx >> 6, e = idx & 63;
        float acc = bias[e];
        for (int ss = 0; ss < nsplit; ++ss)         // fixed order: deterministic
            acc += partials[(size_t)ss * TOKENS * NEXP + gbase + idx];
        s[t * 65 + e] = 1.0f / (1.0f + expf(-acc));
    }
    __syncthreads();

    const int token = blockIdx.x * CHUNK_TOKENS + tid;
    const float* my = &s[tid * 65];
    unsigned long long used = 0ull;

    for (int r = 0; r < TOPK; ++r) {
        float best = -1.0f;   // scores are in (0,1)
        int   bi   = 0;
        for (int e = 0; e < NEXP; ++e) {
            const float v = my[e];
            if (!((used >> e) & 1ull) && v > best) { best = v; bi = e; }
        }
        used |= 1ull << bi;
        sel[(size_t)token * TOPK + r]   = bi;
        selsc[(size_t)token * TOPK + r] = best;
        atomicAdd(&hist[bi], 1);
    }
    __syncthreads();
    if (tid < NEXP) chunkCounts[blockIdx.x * NEXP + tid] = hist[tid];
}

// ---------------------------------------------------------------------------
// Kernel 3: expert totals -> num_tokens_per_expert; exclusive expert offsets;
// per-chunk stable bases. Single block, 64 threads.
// ---------------------------------------------------------------------------
__global__ __launch_bounds__(64) void router_scan_kernel(
    const int* __restrict__ chunkCounts, int* __restrict__ chunkBase,
    int* __restrict__ ntpe_out)
{
    __shared__ int total[NEXP];
    __shared__ int off[NEXP];
    const int e = threadIdx.x;

    int t = 0;
    for (int c = 0; c < NCHUNKS; ++c) t += chunkCounts[c * NEXP + e];
    total[e]    = t;
    ntpe_out[e] = t;
    __syncthreads();
    if (e == 0) {
        int acc = 0;
        for (int i = 0; i < NEXP; ++i) { off[i] = acc; acc += total[i]; }
    }
    __syncthreads();
    int base = off[e];
    for (int c = 0; c < NCHUNKS; ++c) {
        chunkBase[c * NEXP + e] = base;
        base += chunkCounts[c * NEXP + e];
    }
}

// ---------------------------------------------------------------------------
// Kernel 4: stable counting-sort scatter. One block per chunk, one thread per
// expert scans the chunk's 1024 selections in flat order (stable).
// ---------------------------------------------------------------------------
__global__ __launch_bounds__(64) void router_scatter_kernel(
    const int* __restrict__ sel, const float* __restrict__ selsc,
    const int* __restrict__ chunkBase, float* __restrict__ out_sc,
    int* __restrict__ out_tok)
{
    __shared__ int   lsel[CHUNK_TOKENS * TOPK];
    __shared__ float lsc [CHUNK_TOKENS * TOPK];

    const int tid  = threadIdx.x;
    const int base = blockIdx.x * CHUNK_TOKENS * TOPK;
    for (int i = tid; i < CHUNK_TOKENS * TOPK; i += 64) {
        lsel[i] = sel[base + i];
        lsc[i]  = selsc[base + i];
    }
    __syncthreads();

    const int e    = tid;
    int pos        = chunkBase[blockIdx.x * NEXP + e];
    const int tok0 = blockIdx.x * CHUNK_TOKENS;
    for (int i = 0; i < CHUNK_TOKENS * TOPK; ++i) {
        if (lsel[i] == e) {
            out_sc[pos]  = lsc[i];
            out_tok[pos] = tok0 + (i >> 3);
            ++pos;
        }
    }
}

// ---------------------------------------------------------------------------
extern "C" void kernel_launch(void* const* d_in, const int* in_sizes, int n_in,
                              void* d_out, int out_size, void* d_ws, size_t ws_size,
                              hipStream_t stream) {
    const float* x    = (const float*)d_in[0];   // [16384, 2048]
    const float* W    = (const float*)d_in[1];   // [64, 2048]
    const float* bias = (const float*)d_in[2];   // [64]
    (void)in_sizes; (void)n_in; (void)out_size;

    // Pick largest K-split whose partials fit the workspace (ws_size is
    // fixed across calls -> choice is deterministic).
    const size_t tailBytes = (size_t)(2 * NSEL + 2 * NCHUNKS * NEXP) * 4;
    int nsplit = 4;
    while (nsplit > 1 &&
           (size_t)nsplit * TOKENS * NEXP * 4 + tailBytes > ws_size)
        nsplit >>= 1;

    // Workspace layout (all 4-byte elements)
    float* partials    = (float*)d_ws;                             // nsplit*16384*64
    int*   sel         = (int*)(partials + (size_t)nsplit * TOKENS * NEXP); // 131072
    float* selsc       = (float*)(sel + NSEL);                     // 131072
    int*   chunkCounts = (int*)(selsc + NSEL);                     // 128*64
    int*   chunkBase   = chunkCounts + NCHUNKS * NEXP;             // 128*64

    // Output layout: [scores 131072 f32][token_indices 131072 i32][counts 64 i32]
    float* out_sc  = (float*)d_out;
    int*   out_tok = (int*)(out_sc + NSEL);
    int*   ntpe    = (int*)(out_sc + 2 * NSEL);

    const int gemm_blocks = NTILES * nsplit / 8;   // 8 waves per block
    router_gemm_partial_kernel<<<gemm_blocks, 256, 0, stream>>>(x, W, partials, nsplit);
    router_topk_kernel<<<NCHUNKS, 128, 0, stream>>>(partials, bias, sel, selsc,
                                                    chunkCounts, nsplit);
    router_scan_kernel<<<1, 64, 0, stream>>>(chunkCounts, chunkBase, ntpe);
    router_scatter_kernel<<<NCHUNKS, 64, 0, stream>>>(sel, selsc, chunkBase, out_sc, out_tok);
}